// WOneLayerGCN_70162585747786
// MI455X (gfx1250) — compile-verified
//
#include <hip/hip_runtime.h>
#include <hip/hip_bf16.h>

typedef float v2f __attribute__((ext_vector_type(2)));
typedef float v8f __attribute__((ext_vector_type(8)));

// ---------------------------------------------------------------------------
// 1) deg[i] = 1.0 (self-loop weight)
// ---------------------------------------------------------------------------
__global__ void deg_init_kernel(float* __restrict__ deg, int n) {
    int i = blockIdx.x * blockDim.x + threadIdx.x;
    if (i < n) deg[i] = 1.0f;
}

// ---------------------------------------------------------------------------
// 2) deg[col[e]] += w[e]
// ---------------------------------------------------------------------------
__global__ void deg_accum_kernel(const int* __restrict__ col,
                                 const float* __restrict__ w,
                                 float* __restrict__ deg, int e) {
    int i = blockIdx.x * blockDim.x + threadIdx.x;
    if (i < e) atomicAdd(&deg[col[i]], w[i]);
}

// ---------------------------------------------------------------------------
// 3) dis[i] = deg>0 ? rsqrt(deg) : 0   (in place)
// ---------------------------------------------------------------------------
__global__ void deg_to_dis_kernel(float* __restrict__ deg, int n) {
    int i = blockIdx.x * blockDim.x + threadIdx.x;
    if (i < n) {
        float d = deg[i];
        deg[i] = (d > 0.0f) ? rsqrtf(d) : 0.0f;
    }
}

// ---------------------------------------------------------------------------
// 4) h = x @ W via V_WMMA_F32_16X16X4_F32.
//    IN = HID = 128.  Block = 256 threads = 8 waves; block owns one 16-row
//    tile, wave w owns column tile w (8 tiles of 16 = 128 cols).
//    A (16x4 f32): lanes 0-15 = M, v0 holds K0 (lanes<16)/K2 (lanes>=16),
//                  v1 holds K1/K3.   B (4x16): transposed analogue, lanes = N.
//    C/D (16x16 f32): 8 VGPRs; lane<16 -> M=r, lane>=16 -> M=8+r; N = lane&15.
// ---------------------------------------------------------------------------
__global__ __launch_bounds__(256) void gcn_gemm_wmma_kernel(
    const float* __restrict__ x, const float* __restrict__ Wm,
    float* __restrict__ h, int n) {
    const int lane = threadIdx.x & 31;
    const int wave = threadIdx.x >> 5;          // 0..7 -> column tile
    const int row0 = blockIdx.x * 16;
    const int col0 = wave * 16;

    const int hi  = lane >> 4;                  // 0: lanes 0-15, 1: lanes 16-31
    const int l16 = lane & 15;

    int aRow = row0 + l16;
    if (aRow > n - 1) aRow = n - 1;             // clamp loads; EXEC stays full
    const float* xrow = x + (size_t)aRow * 128;
    const int bCol = col0 + l16;

    v8f acc0 = {};
    v8f acc1 = {};
    #pragma unroll
    for (int k = 0; k < 128; k += 8) {
        v2f a0, b0, a1, b1;
        // step k
        a0.x = xrow[k + 2 * hi + 0];
        a0.y = xrow[k + 2 * hi + 1];
        b0.x = Wm[(size_t)(k + 2 * hi + 0) * 128 + bCol];
        b0.y = Wm[(size_t)(k + 2 * hi + 1) * 128 + bCol];
        // step k+4
        a1.x = xrow[k + 4 + 2 * hi + 0];
        a1.y = xrow[k + 4 + 2 * hi + 1];
        b1.x = Wm[(size_t)(k + 4 + 2 * hi + 0) * 128 + bCol];
        b1.y = Wm[(size_t)(k + 4 + 2 * hi + 1) * 128 + bCol];

        acc0 = __builtin_amdgcn_wmma_f32_16x16x4_f32(
            false, a0, false, b0, (short)0, acc0, false, false);
        acc1 = __builtin_amdgcn_wmma_f32_16x16x4_f32(
            false, a1, false, b1, (short)0, acc1, false, false);
    }

    #pragma unroll
    for (int r = 0; r < 8; ++r) {
        int m = row0 + hi * 8 + r;
        if (m < n) h[(size_t)m * 128 + col0 + l16] = acc0[r] + acc1[r];
    }
}

// ---------------------------------------------------------------------------
// 5) out[i][f] = h[i][f] * dis[i]^2 + bias[f]   (self-loop msg + bias)
// ---------------------------------------------------------------------------
__global__ void out_init_kernel(const float* __restrict__ h,
                                const float* __restrict__ dis,
                                const float* __restrict__ bias,
                                float* __restrict__ out, long total) {
    long i = (long)blockIdx.x * blockDim.x + threadIdx.x;
    if (i < total) {
        int node = (int)(i >> 7);
        int f = (int)(i & 127);
        float s = dis[node];
        out[i] = h[i] * s * s + bias[f];
    }
}

// ---------------------------------------------------------------------------
// 6) edge scatter: one wave per edge, lane l handles features [4l, 4l+4)
//    out[col] += h[row] * (dis[row] * w * dis[col])
// ---------------------------------------------------------------------------
__global__ __launch_bounds__(256) void edge_scatter_kernel(
    const int* __restrict__ row, const int* __restrict__ col,
    const float* __restrict__ w, const float* __restrict__ dis,
    const float* __restrict__ h, float* __restrict__ out, int e) {
    long t = (long)blockIdx.x * blockDim.x + threadIdx.x;
    int edge = (int)(t >> 5);
    int lane = (int)(t & 31);
    if (edge >= e) return;

    int r = row[edge];
    int c = col[edge];
    float nrm = dis[r] * w[edge] * dis[c];

    const float4 hv = *(const float4*)(h + (size_t)r * 128 + lane * 4);
    float* op = out + (size_t)c * 128 + lane * 4;
    atomicAdd(op + 0, hv.x * nrm);
    atomicAdd(op + 1, hv.y * nrm);
    atomicAdd(op + 2, hv.z * nrm);
    atomicAdd(op + 3, hv.w * nrm);
}

// ---------------------------------------------------------------------------
// 7) ReLU in place
// ---------------------------------------------------------------------------
__global__ void relu_kernel(float* __restrict__ out, long total) {
    long i = (long)blockIdx.x * blockDim.x + threadIdx.x;
    if (i < total) {
        float v = out[i];
        out[i] = v > 0.0f ? v : 0.0f;
    }
}

// ---------------------------------------------------------------------------
extern "C" void kernel_launch(void* const* d_in, const int* in_sizes, int n_in,
                              void* d_out, int out_size, void* d_ws, size_t ws_size,
                              hipStream_t stream) {
    const float* x          = (const float*)d_in[0];   // [N, IN]
    const int*   edge_index = (const int*)d_in[1];     // [2, E]
    const float* w          = (const float*)d_in[2];   // [E]
    const float* Wm         = (const float*)d_in[3];   // [IN, HID]
    const float* bias       = (const float*)d_in[4];   // [HID]
    float* out = (float*)d_out;                        // [N, HID]

    const int HID = in_sizes[4];                 // 128
    const int IN  = in_sizes[3] / HID;           // 128
    const int n   = in_sizes[0] / IN;            // 50000
    const int e   = in_sizes[2];                 // 800000

    // workspace: dis (n floats) | h (n*HID floats)
    float* dis = (float*)d_ws;
    float* h   = dis + n;

    const int* erow = edge_index;       // sources
    const int* ecol = edge_index + e;   // destinations

    const int T = 256;
    deg_init_kernel<<<(n + T - 1) / T, T, 0, stream>>>(dis, n);
    deg_accum_kernel<<<(e + T - 1) / T, T, 0, stream>>>(ecol, w, dis, e);
    deg_to_dis_kernel<<<(n + T - 1) / T, T, 0, stream>>>(dis, n);

    gcn_gemm_wmma_kernel<<<(n + 15) / 16, T, 0, stream>>>(x, Wm, h, n);

    const long total = (long)n * HID;
    out_init_kernel<<<(int)((total + T - 1) / T), T, 0, stream>>>(h, dis, bias, out, total);

    const long sthreads = (long)e * 32;
    edge_scatter_kernel<<<(int)((sthreads + T - 1) / T), T, 0, stream>>>(
        erow, ecol, w, dis, h, out, e);

    relu_kernel<<<(int)((total + T - 1) / T), T, 0, stream>>>(out, total);
}